// simple_GC_DEC_PyG_80676665688687
// MI455X (gfx1250) — compile-verified
//
#include <hip/hip_runtime.h>
#include <stdint.h>

#define NFEAT 512
#define NHID  64
#define KCL   20
#define ALPHA 0.2f

typedef float v2f __attribute__((ext_vector_type(2)));
typedef float v8f __attribute__((ext_vector_type(8)));

// ---------------- degree / norm ----------------
__global__ void deg_init_kernel(float* __restrict__ deg, int n) {
  int i = blockIdx.x * blockDim.x + threadIdx.x;
  if (i < n) deg[i] = 1.0f;  // self-loop weight
}

__global__ void deg_scatter_kernel(const long long* __restrict__ ei,
                                   const float* __restrict__ w,
                                   float* __restrict__ deg, int e) {
  int i = blockIdx.x * blockDim.x + threadIdx.x;
  if (i < e) {
    long long c = ei[(long long)e + i];  // edge_index[1][i] = target
    atomicAdd(&deg[c], w[i]);
  }
}

__global__ void deg_to_dis_kernel(float* __restrict__ d, int n) {
  int i = blockIdx.x * blockDim.x + threadIdx.x;
  if (i < n) {
    float v = d[i];
    d[i] = (v > 0.0f) ? rsqrtf(fmaxf(v, 1e-30f)) : 0.0f;
  }
}

// ---------------- h = x @ W via V_WMMA_F32_16X16X4_F32 ----------------
// 256 threads = 8 waves; each wave owns a 16-row x 64-col strip.
// W is staged TRANSPOSED (Wt[c][k], stride 68) so each B fragment is one
// contiguous ds_load_b64 into an even-aligned VGPR pair (no shuffle movs),
// with a conflict-free bank pattern (lane c -> banks 4c..4c+1, lane 16+c ->
// banks 4c+2..4c+3).
__global__ __launch_bounds__(256)
void gemm_xw_wmma_kernel(const float* __restrict__ x, const float* __restrict__ W,
                         float* __restrict__ h, int n) {
  constexpr int KT  = 64;        // K-chunk staged per iteration
  constexpr int LDW = KT + 4;    // padded stride (floats)
  __shared__ float Ws[NHID * LDW];  // 64*68*4 = 17.4 KB

  const int lane = threadIdx.x & 31;
  const int wave = threadIdx.x >> 5;  // 0..7
  const int half = lane >> 4;         // 0/1 -> K pair (k,k+1) vs (k+2,k+3)
  const int l16  = lane & 15;
  const int m0   = (blockIdx.x * 8 + wave) * 16;

  // Clamp the A-row address instead of predicating: rows >= n compute garbage
  // accumulators that the guarded store never writes. Keeps EXEC all-1s and
  // removes per-iteration exec save/restore from the hot loop.
  const int row  = m0 + l16;
  const int rowc = (row < n) ? row : (n - 1);
  const float* __restrict__ xrow = x + (size_t)rowc * NFEAT + 2 * half;

  // Per-lane LDS base pointers for the four 16-col B tiles.
  const float* __restrict__ bp0 = Ws + (0 * 16 + l16) * LDW + 2 * half;
  const float* __restrict__ bp1 = Ws + (1 * 16 + l16) * LDW + 2 * half;
  const float* __restrict__ bp2 = Ws + (2 * 16 + l16) * LDW + 2 * half;
  const float* __restrict__ bp3 = Ws + (3 * 16 + l16) * LDW + 2 * half;

  v8f acc0 = {}, acc1 = {}, acc2 = {}, acc3 = {};

  for (int k0 = 0; k0 < NFEAT; k0 += KT) {
    __syncthreads();  // previous tile fully consumed
    // stage W[k0:k0+64, 0:64] transposed -> Ws[c*LDW + k]
    for (int t = threadIdx.x; t < KT * NHID; t += 256) {
      int kk = t >> 6;   // 0..63 (K within chunk)
      int c  = t & 63;   // column; consecutive threads -> coalesced global read
      Ws[c * LDW + kk] = W[(size_t)(k0 + kk) * NHID + c];
    }
    __syncthreads();

    const float* __restrict__ xk = xrow + k0;
    __builtin_prefetch(xk + KT, 0, 0);  // global_prefetch next x chunk

    #pragma unroll 4
    for (int k = 0; k < KT; k += 4) {
      v2f a  = *(const v2f*)(xk + k);    // A 16x4: lanes0-15 K=k,k+1; lanes16-31 K=k+2,k+3
      v2f b0 = *(const v2f*)(bp0 + k);   // one ds_load_b64 each
      v2f b1 = *(const v2f*)(bp1 + k);
      v2f b2 = *(const v2f*)(bp2 + k);
      v2f b3 = *(const v2f*)(bp3 + k);
      acc0 = __builtin_amdgcn_wmma_f32_16x16x4_f32(false, a, false, b0, (short)0, acc0, false, false);
      acc1 = __builtin_amdgcn_wmma_f32_16x16x4_f32(false, a, false, b1, (short)0, acc1, false, false);
      acc2 = __builtin_amdgcn_wmma_f32_16x16x4_f32(false, a, false, b2, (short)0, acc2, false, false);
      acc3 = __builtin_amdgcn_wmma_f32_16x16x4_f32(false, a, false, b3, (short)0, acc3, false, false);
    }
  }

  // C/D layout: VGPR v holds row m0+v (lanes 0-15) / m0+8+v (lanes 16-31), col = l16
  #pragma unroll
  for (int v = 0; v < 8; ++v) {
    const int r = m0 + v + 8 * half;
    if (r < n) {
      float* hr = h + (size_t)r * NHID + l16;
      hr[0]  = acc0[v];
      hr[16] = acc1[v];
      hr[32] = acc2[v];
      hr[48] = acc3[v];
    }
  }
}

// ---------------- out = b + self-loop term (initializes d_out) ----------------
__global__ void out_init_kernel(const float* __restrict__ h, const float* __restrict__ dis,
                                const float* __restrict__ b, float* __restrict__ out, int n) {
  int idx = blockIdx.x * blockDim.x + threadIdx.x;
  if (idx < n * NHID) {
    int i = idx >> 6;
    int j = idx & 63;
    float di = dis[i];
    out[idx] = b[j] + di * di * h[idx];
  }
}

// ---------------- edge message scatter: out[col] += norm * h[row] ----------------
__global__ void edge_scatter_kernel(const long long* __restrict__ ei,
                                    const float* __restrict__ w,
                                    const float* __restrict__ dis,
                                    const float* __restrict__ h,
                                    float* __restrict__ out, int e) {
  long long idx = (long long)blockIdx.x * blockDim.x + threadIdx.x;
  long long eid = idx >> 6;
  int j = (int)(idx & 63);
  if (eid < e) {
    long long r = ei[eid];
    long long c = ei[(long long)e + eid];
    float norm = dis[r] * w[eid] * dis[c];
    atomicAdd(out + c * NHID + j, norm * h[r * NHID + j]);
  }
}

// ---------------- DEC soft assignment ----------------
__global__ __launch_bounds__(256)
void dec_kernel(const float* __restrict__ out, const float* __restrict__ mu,
                float* __restrict__ q, int n) {
  __shared__ float mus[KCL * NHID];
  __shared__ float mu2[KCL];
  for (int t = threadIdx.x; t < KCL * NHID; t += blockDim.x) mus[t] = mu[t];
  __syncthreads();
  if (threadIdx.x < KCL) {
    float s = 0.0f;
    for (int j = 0; j < NHID; ++j) { float v = mus[threadIdx.x * NHID + j]; s += v * v; }
    mu2[threadIdx.x] = s;
  }
  __syncthreads();

  int i = blockIdx.x * blockDim.x + threadIdx.x;
  if (i >= n) return;

  float o[NHID];
  float o2 = 0.0f;
  #pragma unroll
  for (int j = 0; j < NHID; j += 4) {
    float4 t = *(const float4*)(out + (size_t)i * NHID + j);
    o[j] = t.x; o[j + 1] = t.y; o[j + 2] = t.z; o[j + 3] = t.w;
    o2 += t.x * t.x + t.y * t.y + t.z * t.z + t.w * t.w;
  }

  float qs[KCL];
  float ssum = 0.0f;
  #pragma unroll 4
  for (int k = 0; k < KCL; ++k) {
    float dot = 0.0f;
    #pragma unroll
    for (int j = 0; j < NHID; ++j) dot += o[j] * mus[k * NHID + j];
    float sq = o2 + mu2[k] - 2.0f * dot;
    float qq = 1.0f / (1.0f + sq / ALPHA + 1e-8f);
    qq = __powf(qq, ALPHA + 1.0f) * 0.5f;  // (q**1.2)/2 ; /2 cancels in normalize
    qs[k] = qq;
    ssum += qq;
  }
  float inv = 1.0f / ssum;
  #pragma unroll
  for (int k = 0; k < KCL; ++k) q[(size_t)i * KCL + k] = qs[k] * inv;
}

// ---------------- launch ----------------
extern "C" void kernel_launch(void* const* d_in, const int* in_sizes, int n_in,
                              void* d_out, int out_size, void* d_ws, size_t ws_size,
                              hipStream_t stream) {
  (void)n_in; (void)out_size; (void)ws_size;
  const float*     x  = (const float*)d_in[0];      // [N, 512]
  const long long* ei = (const long long*)d_in[1];  // [2, E] int64
  const float*     ea = (const float*)d_in[2];      // [E]
  const float*     W  = (const float*)d_in[3];      // [512, 64]
  const float*     b  = (const float*)d_in[4];      // [64]
  const float*     mu = (const float*)d_in[5];      // [20, 64]

  const int n = in_sizes[0] / NFEAT;  // 100000
  const int e = in_sizes[2];          // 1600000

  float* outp = (float*)d_out;                      // [n, 64]
  float* q    = (float*)d_out + (size_t)n * NHID;   // [n, 20]

  float* dis = (float*)d_ws;        // [n]
  float* h   = dis + n;             // [n, 64]

  const int T = 256;
  deg_init_kernel<<<(n + T - 1) / T, T, 0, stream>>>(dis, n);
  deg_scatter_kernel<<<(e + T - 1) / T, T, 0, stream>>>(ei, ea, dis, e);
  deg_to_dis_kernel<<<(n + T - 1) / T, T, 0, stream>>>(dis, n);

  gemm_xw_wmma_kernel<<<(n + 127) / 128, 256, 0, stream>>>(x, W, h, n);

  out_init_kernel<<<(n * NHID + T - 1) / T, T, 0, stream>>>(h, dis, b, outp, n);

  long long sth = (long long)e * NHID;
  edge_scatter_kernel<<<(unsigned)((sth + T - 1) / T), T, 0, stream>>>(ei, ea, dis, h, outp, e);

  dec_kernel<<<(n + T - 1) / T, T, 0, stream>>>(outp, mu, q, n);
}